// COVID19linear_87909390615037
// MI455X (gfx1250) — compile-verified
//
#include <hip/hip_runtime.h>

typedef __attribute__((ext_vector_type(2))) float v2f;
typedef __attribute__((ext_vector_type(8))) float v8f;

#define N_C    3144      // counties
#define NP     2         // lags
#define NMOB   6
#define NCOV   10
#define T_ALL  156
#define TP     154       // predicted steps = T - P
#define TP_PAD 160       // padded to 10 row tiles of 16
#define NNZ_   31440
#define LDW    3168      // padded leading dim of dense weights: 16*198 >= 3144
#define NWM    ((size_t)N_C * (size_t)LDW)   // elements per dense weight matrix

// workspace layout (floats):
//   WB [N_C x LDW] | WH [N_C x LDW] | WA [N_C x LDW]
// | Csum [TP_PAD x N_C] | Dsum [TP_PAD x N_C]
// | baseC [TP x N_C] | baseD [TP x N_C]
#define OFF_WB   ((size_t)0)
#define OFF_WH   (NWM)
#define OFF_WA   (2 * NWM)
#define OFF_CS   (3 * NWM)
#define OFF_DS   (OFF_CS + (size_t)TP_PAD * N_C)
#define OFF_BC   (OFF_DS + (size_t)TP_PAD * N_C)
#define OFF_BD   (OFF_BC + (size_t)TP * N_C)
#define ZERO_CNT (OFF_BC)   // zero weights + padded Csum/Dsum

__global__ void zero_k(float* __restrict__ p, size_t count) {
    size_t i = (size_t)blockIdx.x * blockDim.x + threadIdx.x;
    size_t stride = (size_t)gridDim.x * blockDim.x;
    for (; i < count; i += stride) p[i] = 0.0f;
}

__global__ void scatter_k(const int* __restrict__ rows, const int* __restrict__ cols,
                          const float* __restrict__ bnz, const float* __restrict__ anz,
                          const float* __restrict__ hnz, float* __restrict__ ws) {
    int i = blockIdx.x * blockDim.x + threadIdx.x;
    if (i >= NNZ_) return;
    size_t off = (size_t)rows[i] * LDW + (size_t)cols[i];
    atomicAdd(ws + OFF_WB + off, bnz[i]);
    atomicAdd(ws + OFF_WA + off, anz[i]);
    atomicAdd(ws + OFF_WH + off, hnz[i]);
}

__global__ void prep_k(const float* __restrict__ C, const float* __restrict__ D,
                       const float* __restrict__ M, const float* __restrict__ cov,
                       const float* __restrict__ mu, const float* __restrict__ nu,
                       const float* __restrict__ ups, const float* __restrict__ zet,
                       float* __restrict__ ws) {
    int idx = blockIdx.x * blockDim.x + threadIdx.x;
    if (idx >= TP * N_C) return;
    int t = idx / N_C;
    int m = idx - t * N_C;

    float cs = C[(size_t)t * N_C + m] + C[(size_t)(t + 1) * N_C + m];
    float dsv = D[(size_t)t * N_C + m] + D[(size_t)(t + 1) * N_C + m];
    ws[OFF_CS + (size_t)t * N_C + m] = cs;
    ws[OFF_DS + (size_t)t * N_C + m] = dsv;

    float mc = 0.0f, md = 0.0f;
#pragma unroll
    for (int k = 0; k < NMOB; ++k) {
#pragma unroll
        for (int tau = 0; tau < NP; ++tau) {
            float mv = M[((size_t)k * T_ALL + (size_t)(t + tau)) * N_C + m];
            mc = fmaf(mv, mu[k * NP + tau], mc);
            md = fmaf(mv, nu[k * NP + tau], md);
        }
    }
    float uc = 0.0f, zc = 0.0f;
#pragma unroll
    for (int j = 0; j < NCOV; ++j) {
        float cv = cov[(size_t)j * N_C + m];
        uc = fmaf(ups[j], cv, uc);
        zc = fmaf(zet[j], cv, zc);
    }
    ws[OFF_BC + (size_t)idx] = mc + uc;
    ws[OFF_BD + (size_t)idx] = md + zc;
}

// One wave computes a 16x32 output slab (two 16x16 WMMA tiles sharing the A frag).
// blockIdx.z == 0: C_hat = Csum @ B + baseC
// blockIdx.z == 1: D_hat = Csum @ H + Dsum @ A + baseD
__global__ __launch_bounds__(128)
void gemm_k(const float* __restrict__ ws, float* __restrict__ out) {
    const float* WB = ws + OFF_WB;
    const float* WH = ws + OFF_WH;
    const float* WA = ws + OFF_WA;
    const float* CS = ws + OFF_CS;
    const float* DS = ws + OFF_DS;

    const int wave = threadIdx.x >> 5;
    const int lane = threadIdx.x & 31;
    const int pair = blockIdx.x * 4 + wave;   // 0..98 cover LDW/32 = 99 col pairs
    if (pair >= 99) return;                    // wave-uniform: EXEC stays all-ones
    const int c0 = pair * 32;                  // columns [c0, c0+32) within LDW
    const int t0 = blockIdx.y * 16;            // row tile (time)
    const int mrow = lane & 15;                // A-frag row / B-frag col within tile
    const int kg = (lane >> 4) * 2;            // lanes 16-31 carry K+2, K+3

    v8f acc0 = {};
    v8f acc1 = {};
    const int z = blockIdx.z;
    const int npass = (z == 0) ? 1 : 2;

    for (int pass = 0; pass < npass; ++pass) {
        const float* X = (z == 0) ? CS : (pass == 0 ? CS : DS);
        const float* W = (z == 0) ? WB : (pass == 0 ? WH : WA);
        const float* xrow = X + (size_t)(t0 + mrow) * N_C;
        for (int k = 0; k < N_C; k += 4) {
            v2f a;
            a.x = xrow[k + kg];
            a.y = xrow[k + kg + 1];
            const float* w0 = W + (size_t)(k + kg) * LDW + (size_t)(c0 + mrow);
            v2f b0, b1;
            b0.x = w0[0];
            b0.y = w0[LDW];
            b1.x = w0[16];
            b1.y = w0[LDW + 16];
            acc0 = __builtin_amdgcn_wmma_f32_16x16x4_f32(false, a, false, b0,
                                                         (short)0, acc0, false, false);
            acc1 = __builtin_amdgcn_wmma_f32_16x16x4_f32(false, a, false, b1,
                                                         (short)0, acc1, false, false);
        }
    }

    const float* base = ws + ((z == 0) ? OFF_BC : OFF_BD);
    const size_t obase = (size_t)z * TP * N_C;
    const int rbump = (lane >> 4) << 3;        // lanes 16-31 hold M = r + 8
#pragma unroll
    for (int r = 0; r < 8; ++r) {
        int row = t0 + r + rbump;
        if (row >= TP) continue;
        int col0 = c0 + mrow;
        if (col0 < N_C)
            out[obase + (size_t)row * N_C + col0] =
                acc0[r] + base[(size_t)row * N_C + col0];
        int col1 = col0 + 16;
        if (col1 < N_C)
            out[obase + (size_t)row * N_C + col1] =
                acc1[r] + base[(size_t)row * N_C + col1];
    }
}

extern "C" void kernel_launch(void* const* d_in, const int* in_sizes, int n_in,
                              void* d_out, int out_size, void* d_ws, size_t ws_size,
                              hipStream_t stream) {
    const float* C   = (const float*)d_in[0];
    const float* D   = (const float*)d_in[1];
    const float* M   = (const float*)d_in[2];
    const float* cov = (const float*)d_in[3];
    const float* bnz = (const float*)d_in[4];
    const float* anz = (const float*)d_in[5];
    const float* hnz = (const float*)d_in[6];
    const float* mu  = (const float*)d_in[7];
    const float* nu  = (const float*)d_in[8];
    const float* ups = (const float*)d_in[9];
    const float* zet = (const float*)d_in[10];
    const int* rows  = (const int*)d_in[11];
    const int* cols  = (const int*)d_in[12];
    float* ws  = (float*)d_ws;
    float* out = (float*)d_out;

    // 1) zero dense weights + padded Csum/Dsum
    zero_k<<<4096, 256, 0, stream>>>(ws, ZERO_CNT);
    // 2) scatter-add nonzeros (duplicates sum)
    scatter_k<<<(NNZ_ + 255) / 256, 256, 0, stream>>>(rows, cols, bnz, anz, hnz, ws);
    // 3) lag fold + mobility/covariate base terms
    prep_k<<<(TP * N_C + 255) / 256, 256, 0, stream>>>(C, D, M, cov, mu, nu, ups, zet, ws);
    // 4) fused WMMA GEMMs + epilogue
    gemm_k<<<dim3(25, TP_PAD / 16, 2), 128, 0, stream>>>(ws, out);
}